// MultiScaleAttention_62663572848961
// MI455X (gfx1250) — compile-verified
//
#include <hip/hip_runtime.h>
#include <hip/hip_bf16.h>
#include <math.h>

// Problem constants (from reference)
#define B_  32
#define S_  2048
#define H_  1024
#define NH_ 16
#define DH_ 64

typedef __attribute__((ext_vector_type(2))) float v2f;
typedef __attribute__((ext_vector_type(8))) float v8f;

// CDNA5 WMMA: D = A(16x4 f32) x B(4x16 f32) + C(16x16 f32), wave32.
__device__ __forceinline__ v8f wmma4(v2f a, v2f b, v8f c) {
  return __builtin_amdgcn_wmma_f32_16x16x4_f32(false, a, false, b, (short)0, c,
                                               false, false);
}

__device__ __forceinline__ float blockReduceSum(float v, float* red) {
  int t = threadIdx.x;
  red[t] = v; __syncthreads();
  for (int s = 128; s > 0; s >>= 1) {
    if (t < s) red[t] += red[t + s];
    __syncthreads();
  }
  float r = red[0]; __syncthreads();
  return r;
}
__device__ __forceinline__ float blockReduceMax(float v, float* red) {
  int t = threadIdx.x;
  red[t] = v; __syncthreads();
  for (int s = 128; s > 0; s >>= 1) {
    if (t < s) red[t] = fmaxf(red[t], red[t + s]);
    __syncthreads();
  }
  float r = red[0]; __syncthreads();
  return r;
}

// ---------------------------------------------------------------------------
// Kt: transpose conv_w (H_out, H_in, 3) into two GEMM-friendly layouts.
//   CW1[o][t*H + i] = conv_w[o][i][t]   (K=o rows,   N=(t,i) cols)  -> A' GEMM
//   CW2[t*H + i][o] = conv_w[o][i][t]   (K=(t,i) rows, N=o cols)    -> ctx_s GEMM
// ---------------------------------------------------------------------------
__global__ void k_transpose_conv(const float* __restrict__ cw,
                                 float* __restrict__ CW1,
                                 float* __restrict__ CW2) {
  int idx = blockIdx.x * 256 + threadIdx.x;
  if (idx >= H_ * H_ * 3) return;
  {
    int o = idx / (3 * H_);
    int rem = idx - o * 3 * H_;
    int t = rem / H_;
    int i = rem - t * H_;
    CW1[idx] = cw[(size_t)o * 3 * H_ + i * 3 + t];
  }
  {
    int k = idx / H_;
    int o = idx - k * H_;
    int t = k / H_;
    int i = k - t * H_;
    CW2[idx] = cw[(size_t)o * 3 * H_ + i * 3 + t];
  }
}

// ---------------------------------------------------------------------------
// K1: q projections of the CLS token for both paths.  QL/QS: [B][H]
// ---------------------------------------------------------------------------
__global__ void k_qproj(const float* __restrict__ hid,
                        const float* __restrict__ Wq_l, const float* __restrict__ bq_l,
                        const float* __restrict__ Wq_s, const float* __restrict__ bq_s,
                        float* __restrict__ QL, float* __restrict__ QS) {
  int b = blockIdx.x, tid = threadIdx.x;
  __shared__ float cls[H_];
  for (int c = tid; c < H_; c += 256) cls[c] = hid[(size_t)b * S_ * H_ + c];
  __syncthreads();
  for (int r = 0; r < 4; r++) {
    int j = tid + r * 256;
    const float4* wl = (const float4*)(Wq_l + (size_t)j * H_);
    const float4* ws = (const float4*)(Wq_s + (size_t)j * H_);
    float al = 0.f, as_ = 0.f;
    for (int c = 0; c < H_ / 4; c++) {
      float4 cv = *(const float4*)(cls + c * 4);
      float4 a = wl[c], s = ws[c];
      al  += cv.x * a.x + cv.y * a.y + cv.z * a.z + cv.w * a.w;
      as_ += cv.x * s.x + cv.y * s.y + cv.z * s.z + cv.w * s.w;
    }
    QL[b * H_ + j] = al + bq_l[j];
    QS[b * H_ + j] = as_ + bq_s[j];
  }
}

// ---------------------------------------------------------------------------
// K2: fold q into Wk:  A_l[b,h,c]=sum_d q_l*Wk_l  (-> P1W rows 0..15)
//                      A_s[b,h,c]               (-> AS buffer)
// score biases: SBL = q_l.bk_l ;  SBS = q_s.bk_s + A_s.conv_b
// ---------------------------------------------------------------------------
__global__ void k_aproj(const float* __restrict__ QL, const float* __restrict__ QS,
                        const float* __restrict__ Wk_l, const float* __restrict__ Wk_s,
                        const float* __restrict__ bk_l, const float* __restrict__ bk_s,
                        const float* __restrict__ conv_b,
                        float* __restrict__ P1W, float* __restrict__ AS,
                        float* __restrict__ SBL, float* __restrict__ SBS) {
  int bh = blockIdx.x, b = bh >> 4, h = bh & 15, tid = threadIdx.x;
  __shared__ float ql[DH_], qs[DH_];
  __shared__ float red[256];
  if (tid < DH_) {
    ql[tid] = QL[b * H_ + h * DH_ + tid];
    qs[tid] = QS[b * H_ + h * DH_ + tid];
  }
  __syncthreads();
  int c0 = tid * 4;
  float aL0 = 0, aL1 = 0, aL2 = 0, aL3 = 0;
  float aS0 = 0, aS1 = 0, aS2 = 0, aS3 = 0;
  for (int d = 0; d < DH_; d++) {
    float4 wl = *(const float4*)(Wk_l + (size_t)(h * DH_ + d) * H_ + c0);
    float4 ws = *(const float4*)(Wk_s + (size_t)(h * DH_ + d) * H_ + c0);
    float qld = ql[d], qsd = qs[d];
    aL0 += qld * wl.x; aL1 += qld * wl.y; aL2 += qld * wl.z; aL3 += qld * wl.w;
    aS0 += qsd * ws.x; aS1 += qsd * ws.y; aS2 += qsd * ws.z; aS3 += qsd * ws.w;
  }
  *(float4*)(P1W + (size_t)b * 64 * H_ + (size_t)h * H_ + c0) =
      make_float4(aL0, aL1, aL2, aL3);
  *(float4*)(AS + (size_t)bh * H_ + c0) = make_float4(aS0, aS1, aS2, aS3);
  float4 cb = *(const float4*)(conv_b + c0);
  float part = aS0 * cb.x + aS1 * cb.y + aS2 * cb.z + aS3 * cb.w;
  float tot = blockReduceSum(part, red);
  if (tid == 0) {
    float sl = 0, ss = 0;
    for (int d = 0; d < DH_; d++) {
      sl += ql[d] * bk_l[h * DH_ + d];
      ss += qs[d] * bk_s[h * DH_ + d];
    }
    SBL[bh] = sl;
    SBS[bh] = ss + tot;
  }
}

// ---------------------------------------------------------------------------
// Generic LDS-tiled WMMA GEMM, C[m][n] = sum_k Arow(m)[k] * B[k][n]
//   mode 0: A' GEMM   (A=AS   rows m*1024,              C into P1W span rows)
//   mode 1: ctx_s GEMM(A=OUT2 span rows (contiguous 3k), C=CTXS rows m*1024)
// Block: 256 thr (8 waves). Tile: M=16 (grid.y), N=128 (grid.x, 16/wave), Kc=32.
// ---------------------------------------------------------------------------
__global__ void k_gemm_wmma(const float* __restrict__ A, const float* __restrict__ Bm,
                            float* __restrict__ C, int K, int ldB, int mode) {
  __shared__ float at[16][36];
  __shared__ float bt[32][132];
  int tid = threadIdx.x, ln = tid & 31, wv = tid >> 5;
  int n0 = blockIdx.x * 128, m0 = blockIdx.y * 16;
  v8f acc = {};
  for (int k0 = 0; k0 < K; k0 += 32) {
    if (tid < 128) {
      int r = tid >> 3, c4 = (tid & 7) * 4;
      int m = m0 + r;
      size_t aro = (mode == 0)
          ? (size_t)m * 1024
          : (size_t)(m >> 4) * 81920 + (size_t)(16 + 3 * (m & 15)) * 1024;
      *(float4*)&at[r][c4] = *(const float4*)(A + aro + k0 + c4);
    }
    for (int t = 0; t < 4; t++) {
      int idx = tid + t * 256;
      int r = idx >> 5, c4 = (idx & 31) * 4;
      *(float4*)&bt[r][c4] = *(const float4*)(Bm + (size_t)(k0 + r) * ldB + n0 + c4);
    }
    __syncthreads();
    int lh = ln & 15;
    #pragma unroll
    for (int kk = 0; kk < 32; kk += 4) {
      int kk2 = kk + 2 * (ln >> 4);
      v2f a, bb;
      a.x = at[lh][kk2];            a.y = at[lh][kk2 + 1];
      bb.x = bt[kk2][wv * 16 + lh]; bb.y = bt[kk2 + 1][wv * 16 + lh];
      acc = wmma4(a, bb, acc);
    }
    __syncthreads();
  }
  int lh = ln & 15;
  #pragma unroll
  for (int v = 0; v < 8; v++) {
    int m = m0 + v + 8 * (ln >> 4);
    size_t cro = (mode == 0)
        ? (size_t)(m >> 4) * 65536 + (size_t)(16 + 3 * (m & 15)) * 1024
        : (size_t)m * 1024;
    C[cro + n0 + wv * 16 + lh] = acc[v];
  }
}

// ---------------------------------------------------------------------------
// P1: streaming scores pass.  SCR[b][n][s] = h[b,s,:].P1W[b][n][:], n=0..63
//   (n<16: local heads; n=16+3h+t: span tap t of head h)
// Block 256 (8 waves): s-tile 128 rows, all 64 cols, K staged by 64 via LDS.
// ---------------------------------------------------------------------------
__global__ void k_scores_wmma(const float* __restrict__ hid,
                              const float* __restrict__ P1W,
                              float* __restrict__ SCR) {
  __shared__ float hs[128][68];
  __shared__ float wt[64][68];
  int b = blockIdx.y, s0 = blockIdx.x * 128;
  int tid = threadIdx.x, ln = tid & 31, wv = tid >> 5;
  const float* hb = hid + (size_t)b * S_ * H_;
  const float* wb = P1W + (size_t)b * 64 * H_;
  v8f acc[4];
  { v8f z = {}; for (int j = 0; j < 4; j++) acc[j] = z; }
  for (int kb = 0; kb < H_; kb += 64) {
    for (int t = 0; t < 8; t++) {
      int idx = tid + t * 256;                 // 128 rows x 16 float4
      int r = idx >> 4, c4 = (idx & 15) * 4;
      *(float4*)&hs[r][c4] = *(const float4*)(hb + (size_t)(s0 + r) * H_ + kb + c4);
    }
    for (int t = 0; t < 4; t++) {
      int idx = tid + t * 256;                 // 64 rows x 16 float4
      int r = idx >> 4, c4 = (idx & 15) * 4;
      *(float4*)&wt[r][c4] = *(const float4*)(wb + (size_t)r * H_ + kb + c4);
    }
    __syncthreads();
    int lh = ln & 15, wrow = wv * 16;
    #pragma unroll
    for (int kk = 0; kk < 64; kk += 4) {
      int kk2 = kk + 2 * (ln >> 4);
      v2f a;
      a.x = hs[wrow + lh][kk2]; a.y = hs[wrow + lh][kk2 + 1];
      #pragma unroll
      for (int j = 0; j < 4; j++) {
        v2f bb;
        bb.x = wt[j * 16 + lh][kk2]; bb.y = wt[j * 16 + lh][kk2 + 1];
        acc[j] = wmma4(a, bb, acc[j]);
      }
    }
    __syncthreads();
  }
  int lh = ln & 15;
  #pragma unroll
  for (int j = 0; j < 4; j++)
    #pragma unroll
    for (int v = 0; v < 8; v++) {
      int n = j * 16 + lh;
      int s = s0 + wv * 16 + v + 8 * (ln >> 4);
      SCR[((size_t)b * 64 + n) * S_ + s] = acc[j][v];
    }
}

// ---------------------------------------------------------------------------
// K5: masked softmax over S per (b, head) + build pass-2 weight matrix.
//  WGT[b][m][s]: m<16 local w ; m=16+3h+t shifted span w ; m=64 global mean w ;
//  m=65..79 zero pad.
// ---------------------------------------------------------------------------
__global__ void k_softmax(const float* __restrict__ SCR, const float* __restrict__ SBL,
                          const float* __restrict__ SBS, const int* __restrict__ amask,
                          float* __restrict__ WGT) {
  int b = blockIdx.y, gx = blockIdx.x, tid = threadIdx.x;
  __shared__ float red[256];
  const int* mrow = amask + (size_t)b * S_;
  if (gx < 32) {
    bool isLocal = gx < 16;
    int h = isLocal ? gx : gx - 16;
    float sb = isLocal ? SBL[b * 16 + h] : SBS[b * 16 + h];
    float xv[8];
    float mx = -3.4e38f;
    if (isLocal) {
      const float* q1 = SCR + ((size_t)b * 64 + h) * S_;
      for (int i = 0; i < 8; i++) {
        int s = tid + i * 256;
        float x = (q1[s] + sb) * 0.125f;
        if (mrow[s] == 0) x = -1.0e9f;
        xv[i] = x; mx = fmaxf(mx, x);
      }
    } else {
      const float* q0 = SCR + ((size_t)b * 64 + 16 + 3 * h + 0) * S_;
      const float* q1 = SCR + ((size_t)b * 64 + 16 + 3 * h + 1) * S_;
      const float* q2 = SCR + ((size_t)b * 64 + 16 + 3 * h + 2) * S_;
      for (int i = 0; i < 8; i++) {
        int s = tid + i * 256;
        float x = q1[s];
        if (s > 0) x += q0[s - 1];          // conv zero padding at edges
        if (s < S_ - 1) x += q2[s + 1];
        x = (x + sb) * 0.125f;
        if (mrow[s] == 0) x = -1.0e9f;
        xv[i] = x; mx = fmaxf(mx, x);
      }
    }
    mx = blockReduceMax(mx, red);
    float sum = 0.f;
    for (int i = 0; i < 8; i++) { float e = expf(xv[i] - mx); xv[i] = e; sum += e; }
    sum = blockReduceSum(sum, red);
    float inv = 1.0f / sum;
    if (isLocal) {
      float* r1 = WGT + ((size_t)b * 80 + h) * S_;
      for (int i = 0; i < 8; i++) { int s = tid + i * 256; r1[s] = xv[i] * inv; }
    } else {
      float* r0 = WGT + ((size_t)b * 80 + 16 + 3 * h + 0) * S_;
      float* r1 = WGT + ((size_t)b * 80 + 16 + 3 * h + 1) * S_;
      float* r2 = WGT + ((size_t)b * 80 + 16 + 3 * h + 2) * S_;
      for (int i = 0; i < 8; i++) {
        int s = tid + i * 256;
        float w = xv[i] * inv;
        r1[s] = w;
        if (s > 0) r0[s - 1] = w;
        if (s < S_ - 1) r2[s + 1] = w;
        if (s == 0) r2[0] = 0.0f;
        if (s == S_ - 1) r0[S_ - 1] = 0.0f;
      }
    }
  } else {
    float cnt = 0.f;
    for (int i = 0; i < 8; i++) { int s = tid + i * 256; cnt += (mrow[s] != 0) ? 1.f : 0.f; }
    cnt = blockReduceSum(cnt, red);
    float inv = 1.0f / fmaxf(cnt, 1.0f);
    float* rg = WGT + ((size_t)b * 80 + 64) * S_;
    for (int i = 0; i < 8; i++) { int s = tid + i * 256; rg[s] = (mrow[s] != 0) ? inv : 0.f; }
    float* pz = WGT + ((size_t)b * 80 + 65) * S_;
    for (int idx = tid; idx < 15 * S_; idx += 256) pz[idx] = 0.0f;
  }
}

// ---------------------------------------------------------------------------
// P2: streaming context pass. OUT2[b][m][c] = sum_s WGT[b][m][s] * h[b][s][c]
// Block 256 (8 waves): all 80 m-rows (5 tiles/wave), 128 c-cols, S staged by 32.
// ---------------------------------------------------------------------------
__global__ void k_ctx_wmma(const float* __restrict__ hid,
                           const float* __restrict__ WGT,
                           float* __restrict__ OUT2) {
  __shared__ float ht[32][132];
  __shared__ float wg[80][36];
  int b = blockIdx.y, cb = blockIdx.x * 128;
  int tid = threadIdx.x, ln = tid & 31, wv = tid >> 5;
  const float* hb = hid + (size_t)b * S_ * H_;
  const float* gb = WGT + (size_t)b * 80 * S_;
  v8f acc[5];
  { v8f z = {}; for (int j = 0; j < 5; j++) acc[j] = z; }
  for (int s0 = 0; s0 < S_; s0 += 32) {
    for (int t = 0; t < 4; t++) {
      int idx = tid + t * 256;                 // 32 rows x 32 float4
      int r = idx >> 5, c4 = (idx & 31) * 4;
      *(float4*)&ht[r][c4] = *(const float4*)(hb + (size_t)(s0 + r) * H_ + cb + c4);
    }
    {
      int r0 = tid >> 3, c4 = (tid & 7) * 4;   // 80 rows x 8 float4
      *(float4*)&wg[r0][c4]      = *(const float4*)(gb + (size_t)r0 * S_ + s0 + c4);
      *(float4*)&wg[r0 + 32][c4] = *(const float4*)(gb + (size_t)(r0 + 32) * S_ + s0 + c4);
      if (r0 < 16)
        *(float4*)&wg[r0 + 64][c4] = *(const float4*)(gb + (size_t)(r0 + 64) * S_ + s0 + c4);
    }
    __syncthreads();
    int lh = ln & 15;
    #pragma unroll
    for (int kk = 0; kk < 32; kk += 4) {
      int kk2 = kk + 2 * (ln >> 4);
      v2f bb;
      bb.x = ht[kk2][wv * 16 + lh]; bb.y = ht[kk2 + 1][wv * 16 + lh];
      #pragma unroll
      for (int mt = 0; mt < 5; mt++) {
        v2f a;
        a.x = wg[mt * 16 + lh][kk2]; a.y = wg[mt * 16 + lh][kk2 + 1];
        acc[mt] = wmma4(a, bb, acc[mt]);
      }
    }
    __syncthreads();
  }
  int lh = ln & 15;
  #pragma unroll
  for (int mt = 0; mt < 5; mt++)
    #pragma unroll
    for (int v = 0; v < 8; v++) {
      int m = mt * 16 + v + 8 * (ln >> 4);
      OUT2[((size_t)b * 80 + m) * H_ + cb + wv * 16 + lh] = acc[mt][v];
    }
}

// ---------------------------------------------------------------------------
// K7b: per-head value projection of the context vectors.
//   o_l[b][j] = ctx_l[b, j>>6, :].Wv_l[j,:] + bv_l[j]
//   o_s uses (ctx_s + conv_b).
// ---------------------------------------------------------------------------
__global__ void k_vproj(const float* __restrict__ OUT2, const float* __restrict__ CTXS,
                        const float* __restrict__ conv_b,
                        const float* __restrict__ Wv_l, const float* __restrict__ bv_l,
                        const float* __restrict__ Wv_s, const float* __restrict__ bv_s,
                        float* __restrict__ OL, float* __restrict__ OS) {
  int b = blockIdx.x, tid = threadIdx.x;
  for (int r = 0; r < 4; r++) {
    int j = tid + r * 256;
    int h = j >> 6;
    const float4* cl  = (const float4*)(OUT2 + ((size_t)b * 80 + h) * H_);
    const float4* cs  = (const float4*)(CTXS + (size_t)(b * 16 + h) * H_);
    const float4* cb4 = (const float4*)conv_b;
    const float4* wl  = (const float4*)(Wv_l + (size_t)j * H_);
    const float4* ws  = (const float4*)(Wv_s + (size_t)j * H_);
    float al = 0.f, as_ = 0.f;
    for (int c = 0; c < H_ / 4; c++) {
      float4 x = cl[c], wlv = wl[c];
      al += x.x * wlv.x + x.y * wlv.y + x.z * wlv.z + x.w * wlv.w;
      float4 y = cs[c], bb = cb4[c], wsv = ws[c];
      as_ += (y.x + bb.x) * wsv.x + (y.y + bb.y) * wsv.y +
             (y.z + bb.z) * wsv.z + (y.w + bb.w) * wsv.w;
    }
    OL[b * H_ + j] = al + bv_l[j];
    OS[b * H_ + j] = as_ + bv_s[j];
  }
}

// ---------------------------------------------------------------------------
// K7c: output projections -> FUSED[b][3H] = [local_out | span_out | global_out]
// ---------------------------------------------------------------------------
__global__ void k_oproj(const float* __restrict__ OL, const float* __restrict__ OS,
                        const float* __restrict__ OUT2,
                        const float* __restrict__ Wo_l, const float* __restrict__ bo_l,
                        const float* __restrict__ Wo_s, const float* __restrict__ bo_s,
                        float* __restrict__ FUSED) {
  int b = blockIdx.x, tid = threadIdx.x;
  __shared__ float ol[H_], os[H_];
  for (int c = tid; c < H_; c += 256) { ol[c] = OL[b * H_ + c]; os[c] = OS[b * H_ + c]; }
  __syncthreads();
  for (int r = 0; r < 4; r++) {
    int j = tid + r * 256;
    const float4* wl = (const float4*)(Wo_l + (size_t)j * H_);
    const float4* ws = (const float4*)(Wo_s + (size_t)j * H_);
    float al = 0.f, as_ = 0.f;
    for (int c = 0; c < H_ / 4; c++) {
      float4 xo = *(const float4*)(ol + c * 4); float4 wv_ = wl[c];
      al += xo.x * wv_.x + xo.y * wv_.y + xo.z * wv_.z + xo.w * wv_.w;
      float4 ys = *(const float4*)(os + c * 4); float4 wsv = ws[c];
      as_ += ys.x * wsv.x + ys.y * wsv.y + ys.z * wsv.z + ys.w * wsv.w;
    }
    FUSED[(size_t)b * 3 * H_ + j]            = al + bo_l[j];
    FUSED[(size_t)b * 3 * H_ + H_ + j]       = as_ + bo_s[j];
    FUSED[(size_t)b * 3 * H_ + 2 * H_ + j]   = OUT2[((size_t)b * 80 + 64) * H_ + j];
  }
}

// ---------------------------------------------------------------------------
// K7d: fusion projection (Wf) + layernorm -> final output [B][H]
// ---------------------------------------------------------------------------
__global__ void k_fuse_ln(const float* __restrict__ FUSED, const float* __restrict__ Wf,
                          const float* __restrict__ bf, const float* __restrict__ gamma,
                          const float* __restrict__ beta, float* __restrict__ out) {
  int b = blockIdx.x, tid = threadIdx.x;
  __shared__ float fz[3 * H_];
  __shared__ float red[256];
  for (int c = tid; c < 3 * H_; c += 256) fz[c] = FUSED[(size_t)b * 3 * H_ + c];
  __syncthreads();
  float yloc[4];
  float part = 0.f;
  for (int r = 0; r < 4; r++) {
    int j = tid + r * 256;
    const float4* wf = (const float4*)(Wf + (size_t)j * 3 * H_);
    float acc = 0.f;
    for (int c = 0; c < 3 * H_ / 4; c++) {
      float4 f = *(const float4*)(fz + c * 4);
      float4 w = wf[c];
      acc += f.x * w.x + f.y * w.y + f.z * w.z + f.w * w.w;
    }
    acc += bf[j];
    yloc[r] = acc; part += acc;
  }
  float mu = blockReduceSum(part, red) / (float)H_;
  float vpart = 0.f;
  for (int r = 0; r < 4; r++) { float d = yloc[r] - mu; vpart += d * d; }
  float var = blockReduceSum(vpart, red) / (float)H_;
  float inv = rsqrtf(var + 1e-5f);
  for (int r = 0; r < 4; r++) {
    int j = tid + r * 256;
    out[(size_t)b * H_ + j] = (yloc[r] - mu) * inv * gamma[j] + beta[j];
  }
}

// ---------------------------------------------------------------------------
extern "C" void kernel_launch(void* const* d_in, const int* in_sizes, int n_in,
                              void* d_out, int out_size, void* d_ws, size_t ws_size,
                              hipStream_t stream) {
  const float* hid    = (const float*)d_in[0];
  const int*   amask  = (const int*)d_in[1];
  const float* Wq_l = (const float*)d_in[2],  *Wk_l = (const float*)d_in[3];
  const float* Wv_l = (const float*)d_in[4],  *Wo_l = (const float*)d_in[5];
  const float* bq_l = (const float*)d_in[6],  *bk_l = (const float*)d_in[7];
  const float* bv_l = (const float*)d_in[8],  *bo_l = (const float*)d_in[9];
  const float* Wq_s = (const float*)d_in[10], *Wk_s = (const float*)d_in[11];
  const float* Wv_s = (const float*)d_in[12], *Wo_s = (const float*)d_in[13];
  const float* bq_s = (const float*)d_in[14], *bk_s = (const float*)d_in[15];
  const float* bv_s = (const float*)d_in[16], *bo_s = (const float*)d_in[17];
  const float* conv_w = (const float*)d_in[18], *conv_b = (const float*)d_in[19];
  const float* Wf = (const float*)d_in[20], *bf = (const float*)d_in[21];
  const float* gamma = (const float*)d_in[22], *beta = (const float*)d_in[23];
  float* out = (float*)d_out;

  // Workspace carve-up (floats)
  float* ws = (float*)d_ws;
  size_t off = 0;
  float* QL    = ws + off; off += (size_t)B_ * H_;             // 32K
  float* QS    = ws + off; off += (size_t)B_ * H_;
  float* P1W   = ws + off; off += (size_t)B_ * 64 * H_;        // pass-1 weights
  float* AS    = ws + off; off += (size_t)B_ * NH_ * H_;       // A_s
  float* SBL   = ws + off; off += (size_t)B_ * NH_;
  float* SBS   = ws + off; off += (size_t)B_ * NH_;
  float* CW1   = ws + off; off += (size_t)H_ * 3 * H_;         // conv_w transpose 1
  float* CW2   = ws + off; off += (size_t)3 * H_ * H_;         // conv_w transpose 2
  float* SCR   = ws + off; off += (size_t)B_ * 64 * S_;        // raw scores
  float* WGT   = ws + off; off += (size_t)B_ * 80 * S_;        // softmax weight rows
  float* OUT2  = ws + off; off += (size_t)B_ * 80 * H_;        // contexts
  float* CTXS  = ws + off; off += (size_t)B_ * NH_ * H_;       // span ctx
  float* OL    = ws + off; off += (size_t)B_ * H_;
  float* OS    = ws + off; off += (size_t)B_ * H_;
  float* FUSED = ws + off; off += (size_t)B_ * 3 * H_;
  (void)ws_size; (void)in_sizes; (void)n_in; (void)out_size;

  // 1. conv_w transposes
  k_transpose_conv<<<dim3((H_ * H_ * 3 + 255) / 256), 256, 0, stream>>>(conv_w, CW1, CW2);
  // 2. q projections
  k_qproj<<<dim3(B_), 256, 0, stream>>>(hid, Wq_l, bq_l, Wq_s, bq_s, QL, QS);
  // 3. fold q into Wk (A_l -> P1W rows 0..15, A_s, score biases)
  k_aproj<<<dim3(B_ * NH_), 256, 0, stream>>>(QL, QS, Wk_l, Wk_s, bk_l, bk_s,
                                              conv_b, P1W, AS, SBL, SBS);
  // 4. A' = A_s x conv_w  -> P1W rows 16..63  (M=512, K=1024, N=3072)
  k_gemm_wmma<<<dim3(3 * H_ / 128, B_ * NH_ / 16), 256, 0, stream>>>(
      AS, CW1, P1W, /*K=*/H_, /*ldB=*/3 * H_, /*mode=*/0);
  // 5. streaming scores pass (WMMA, reads hidden once)
  k_scores_wmma<<<dim3(S_ / 128, B_), 256, 0, stream>>>(hid, P1W, SCR);
  // 6. masked softmax + weight-row construction
  k_softmax<<<dim3(33, B_), 256, 0, stream>>>(SCR, SBL, SBS, amask, WGT);
  // 7. streaming context pass (WMMA, reads hidden once)
  k_ctx_wmma<<<dim3(H_ / 128, B_), 256, 0, stream>>>(hid, WGT, OUT2);
  // 8. ctx_s = g x conv_w^T  (M=512, K=3072, N=1024)
  k_gemm_wmma<<<dim3(H_ / 128, B_ * NH_ / 16), 256, 0, stream>>>(
      OUT2, CW2, CTXS, /*K=*/3 * H_, /*ldB=*/H_, /*mode=*/1);
  // 9. value projections
  k_vproj<<<dim3(B_), 256, 0, stream>>>(OUT2, CTXS, conv_b, Wv_l, bv_l, Wv_s, bv_s, OL, OS);
  // 10. output projections + fuse
  k_oproj<<<dim3(B_), 256, 0, stream>>>(OL, OS, OUT2, Wo_l, bo_l, Wo_s, bo_s, FUSED);
  // 11. Wf + layernorm
  k_fuse_ln<<<dim3(B_), 256, 0, stream>>>(FUSED, Wf, bf, gamma, beta, out);
}